// GCNLayer_16449724744840
// MI455X (gfx1250) — compile-verified
//
#include <hip/hip_runtime.h>
#include <hip/hip_bf16.h>

#define N_NODES 40000
#define N_EDGES 640000
#define D 128
#define LDS_STRIDE 132   // 128 + 4 pad: stride mod 64 banks = 4 -> conflict-free column reads

typedef __attribute__((ext_vector_type(2))) float v2f;
typedef __attribute__((ext_vector_type(8))) float v8f;

// ---------------------------------------------------------------------------
// Kernel 1: zero the accumulator (d_out doubles as h).
// ---------------------------------------------------------------------------
__global__ void gcn_zero_kernel(float4* __restrict__ p, int n4) {
    int i = blockIdx.x * blockDim.x + threadIdx.x;
    if (i < n4) p[i] = make_float4(0.f, 0.f, 0.f, 0.f);
}

// ---------------------------------------------------------------------------
// Kernel 2: edge gather + atomic scatter-add.
// One wave32 per edge: lane handles 4 consecutive features (float4 load,
// 4x global_atomic_add_f32). h (= d_out) is 20.5MB -> L2 resident.
// ---------------------------------------------------------------------------
__global__ void gcn_scatter_kernel(const float* __restrict__ x,
                                   const int* __restrict__ src,
                                   const int* __restrict__ dst,
                                   float* __restrict__ h) {
    int g    = blockIdx.x * blockDim.x + threadIdx.x;
    int e    = g >> 5;
    int lane = g & 31;
    if (e >= N_EDGES) return;
    int s = src[e];
    int d = dst[e];
    const float4 v = *(const float4*)(x + (size_t)s * D + lane * 4);
    float* hp = h + (size_t)d * D + lane * 4;
    atomicAdd(hp + 0, v.x);
    atomicAdd(hp + 1, v.y);
    atomicAdd(hp + 2, v.z);
    atomicAdd(hp + 3, v.w);
}

// ---------------------------------------------------------------------------
// Kernel 3: out = h @ W^T + b, in place on d_out, using V_WMMA_F32_16X16X4_F32.
// Block = 256 threads = 8 wave32. Block owns 16 output rows; wave w owns the
// 16-column tile starting at 16*w. K=128 accumulated over 32 WMMA(K=4) steps.
//
// Fragment layouts (CDNA5 ISA 7.12.2):
//   A 16x4 f32 : lane L -> M = L%16, VGPR0/1 hold K = 2*(L/16) + {0,1}
//   B 4x16 f32 : lane L -> N = L%16, VGPR0/1 hold K = 2*(L/16) + {0,1}
//   C/D 16x16  : VGPR v -> M = v + 8*(L/16), N = L%16
// ---------------------------------------------------------------------------
__global__ void __launch_bounds__(256)
gcn_gemm_wmma_kernel(float* __restrict__ hout,          // in: h, out: result
                     const float* __restrict__ W,       // [D_OUT=128, D_IN=128] row-major
                     const float* __restrict__ bias) {  // [128]
    __shared__ float Wl[D * LDS_STRIDE];   // whole W, padded   (67.6 KB)
    __shared__ float Hl[16 * LDS_STRIDE];  // 16-row h tile      (8.4 KB)

    const int tid = threadIdx.x;
    const int m0  = blockIdx.x * 16;       // 40000/16 = 2500 blocks, exact

    // Stage full W into LDS (128x128), padded rows.
    for (int i = tid; i < D * D; i += 256) {
        int r = i >> 7, c = i & (D - 1);
        Wl[r * LDS_STRIDE + c] = W[i];
    }
    // Stage this block's 16x128 h tile.
    for (int i = tid; i < 16 * D; i += 256) {
        int r = i >> 7, c = i & (D - 1);
        Hl[r * LDS_STRIDE + c] = hout[(size_t)(m0 + r) * D + c];
    }
    __syncthreads();

    const int wave = tid >> 5;             // 0..7 -> column tile
    const int lane = tid & 31;
    const int half = lane >> 4;            // 0 or 1
    const int mn   = lane & 15;            // M for A, N for B
    const int n0   = wave * 16;

    v8f acc = {};
    #pragma unroll
    for (int kk = 0; kk < D; kk += 4) {
        const int kb = kk + 2 * half;      // even -> 8B-aligned LDS reads
        v2f a = *(const v2f*)(&Hl[mn * LDS_STRIDE + kb]);
        v2f b = *(const v2f*)(&Wl[(n0 + mn) * LDS_STRIDE + kb]);
        acc = __builtin_amdgcn_wmma_f32_16x16x4_f32(
            /*neg_a=*/false, a, /*neg_b=*/false, b,
            /*c_mod=*/(short)0, acc, /*reuse_a=*/false, /*reuse_b=*/false);
    }

    const float bv = bias[n0 + mn];
    #pragma unroll
    for (int v = 0; v < 8; ++v) {
        int row = v + 8 * half;
        hout[(size_t)(m0 + row) * D + n0 + mn] = acc[v] + bv;
    }
}

// ---------------------------------------------------------------------------
extern "C" void kernel_launch(void* const* d_in, const int* in_sizes, int n_in,
                              void* d_out, int out_size, void* d_ws, size_t ws_size,
                              hipStream_t stream) {
    const float* x   = (const float*)d_in[0];   // [40000,128]
    const int*   src = (const int*)  d_in[1];   // [640000]
    const int*   dst = (const int*)  d_in[2];   // [640000]
    const float* W   = (const float*)d_in[3];   // [128,128]
    const float* b   = (const float*)d_in[4];   // [128]
    float* out = (float*)d_out;                 // [40000,128]

    // 1) zero accumulator (d_out reused as h)
    {
        int n4 = (N_NODES * D) / 4;             // 1,280,000
        gcn_zero_kernel<<<(n4 + 255) / 256, 256, 0, stream>>>((float4*)out, n4);
    }
    // 2) scatter-add edges: one wave per edge
    {
        long long threads = (long long)N_EDGES * 32;
        gcn_scatter_kernel<<<(int)(threads / 256), 256, 0, stream>>>(x, src, dst, out);
    }
    // 3) in-place WMMA GEMM + bias
    gcn_gemm_wmma_kernel<<<N_NODES / 16, 256, 0, stream>>>(out, W, b);
}